// PointNetPolylineEncoder_72516227826271
// MI455X (gfx1250) — compile-verified
//
#include <hip/hip_runtime.h>
#include <hip/hip_bf16.h>

// PointNet polyline encoder for MI455X (gfx1250), fp32 WMMA path.
// B=32 P=768 N=20 C=9 H=64 O=128. One WG = 8 polylines = 160 rows = 10 M-tiles.

typedef float v2f __attribute__((ext_vector_type(2)));
typedef float v8f __attribute__((ext_vector_type(8)));

#define BB   32
#define PP   768
#define NPT  20
#define CC   9
#define HH   64
#define OO   128
#define PG   8
#define ROWS (PG * NPT)      // 160
#define LDF  68              // padded feature stride (floats): conflict-free ds_load_b64
#define LDW1 132             // padded W1 stride (K=128 + 4)
#define BN_EPS 1e-5f

// ---- dynamic-LDS float offsets ----
#define OFF_F1   0
#define OFF_F2   (OFF_F1 + ROWS * LDF)        // layer outputs ping-pong (sX staged here too)
#define OFF_P1   (OFF_F2 + ROWS * LDF)        // pooled1: 8 x LDF
#define OFF_P2   (OFF_P1 + PG * LDF)          // pooled2 padded to 16 rows
#define OFF_HID  (OFF_P2 + 16 * LDF)          // head hidden: 16 x LDF
#define OFF_W    (OFF_HID + 16 * LDF)         // reused weight buffer, 8704 floats
#define OFF_W0T  (OFF_W + 8704)               // W0 transposed [c][h]
#define OFF_SC0  (OFF_W0T + CC * HH)
#define OFF_SH0  (OFF_SC0 + HH)
#define OFF_SC1  (OFF_SH0 + HH)
#define OFF_SH1  (OFF_SC1 + HH)
#define OFF_SC2  (OFF_SH1 + HH)
#define OFF_SH2  (OFF_SC2 + HH)
#define OFF_BO1  (OFF_SH2 + HH)
#define OFF_BO2  (OFF_BO1 + HH)
#define OFF_MSK  (OFF_BO2 + OO)
#define OFF_VAL  (OFF_MSK + ROWS)
#define SMEM_FLOATS (OFF_VAL + PG)            // 34504 floats = 138016 bytes (< 320KB WGP LDS)

__device__ __forceinline__ v8f wmma_f32(v2f a, v2f b, v8f c) {
  return __builtin_amdgcn_wmma_f32_16x16x4_f32(false, a, false, b, (short)0, c,
                                               false, false);
}

__global__ __launch_bounds__(256)
void pointnet_poly_kernel(const float* __restrict__ poly,
                          const unsigned char* __restrict__ maskb,
                          const float* __restrict__ W0, const float* __restrict__ g0,
                          const float* __restrict__ b0, const float* __restrict__ m0,
                          const float* __restrict__ v0,
                          const float* __restrict__ W1, const float* __restrict__ g1,
                          const float* __restrict__ b1, const float* __restrict__ m1,
                          const float* __restrict__ v1,
                          const float* __restrict__ W2, const float* __restrict__ g2,
                          const float* __restrict__ b2, const float* __restrict__ m2,
                          const float* __restrict__ v2,
                          const float* __restrict__ Wo1, const float* __restrict__ bo1,
                          const float* __restrict__ Wo2, const float* __restrict__ bo2,
                          float* __restrict__ out) {
  extern __shared__ float sm[];
  float* sF1  = sm + OFF_F1;
  float* sF2  = sm + OFF_F2;
  float* sX   = sF2;            // input staging alias; dead before sF2 is written
  float* sP1  = sm + OFF_P1;
  float* sP2  = sm + OFF_P2;
  float* sHid = sm + OFF_HID;
  float* sW   = sm + OFF_W;
  float* sW0T = sm + OFF_W0T;
  float* sSc0 = sm + OFF_SC0;  float* sSh0 = sm + OFF_SH0;
  float* sSc1 = sm + OFF_SC1;  float* sSh1 = sm + OFF_SH1;
  float* sSc2 = sm + OFF_SC2;  float* sSh2 = sm + OFF_SH2;
  float* sBo1 = sm + OFF_BO1;  float* sBo2 = sm + OFF_BO2;
  float* sMask = sm + OFF_MSK; float* sValid = sm + OFF_VAL;

  const int tid  = threadIdx.x;
  const int wave = tid >> 5;
  const int lane = tid & 31;
  const int half = lane >> 4;   // 0: lanes 0-15, 1: lanes 16-31
  const int lq   = lane & 15;
  const int q0   = blockIdx.x * PG;   // first polyline of this WG

  // ---------- Phase A: stage inputs, W1, BN folds ----------
  {
    const float* gx = poly + (size_t)q0 * NPT * CC;     // 160*9 contiguous floats
    for (int e = tid; e < ROWS * CC; e += 256) {
      int r = e / CC, c = e - r * CC;
      sX[r * 12 + c] = gx[e];
    }
    const unsigned char* gm = maskb + (size_t)q0 * NPT;
    for (int e = tid; e < ROWS; e += 256) sMask[e] = gm[e] ? 1.f : 0.f;
    for (int e = tid; e < HH * CC; e += 256) {
      int h = e / CC, c = e - h * CC;
      sW0T[c * HH + h] = W0[e];
    }
    for (int e = tid; e < HH * 2 * HH; e += 256) {      // W1: 64x128
      int h = e >> 7, k = e & 127;
      sW[h * LDW1 + k] = W1[e];
    }
    if (tid < HH) {
      float s0 = g0[tid] * rsqrtf(v0[tid] + BN_EPS);
      sSc0[tid] = s0; sSh0[tid] = b0[tid] - m0[tid] * s0;
      float s1 = g1[tid] * rsqrtf(v1[tid] + BN_EPS);
      sSc1[tid] = s1; sSh1[tid] = b1[tid] - m1[tid] * s1;
      float s2 = g2[tid] * rsqrtf(v2[tid] + BN_EPS);
      sSc2[tid] = s2; sSh2[tid] = b2[tid] - m2[tid] * s2;
      sBo1[tid] = bo1[tid];
    }
    if (tid < OO) sBo2[tid] = bo2[tid];
  }
  __syncthreads();

  // ---------- Phase B: layer0 (K=9) on VALU, BN+ReLU+mask ----------
  for (int o = tid; o < ROWS * HH; o += 256) {
    int r = o >> 6, h = o & 63;
    float acc = 0.f;
#pragma unroll
    for (int c = 0; c < CC; ++c) acc += sX[r * 12 + c] * sW0T[c * HH + h];
    sF1[r * LDF + h] = fmaxf(acc * sSc0[h] + sSh0[h], 0.f) * sMask[r];
  }
  __syncthreads();

  // ---------- Phase C: max-pool over points + polyline validity ----------
  for (int o = tid; o < PG * HH; o += 256) {
    int pl = o >> 6, h = o & 63;
    float mx = 0.f;
    for (int n = 0; n < NPT; ++n) mx = fmaxf(mx, sF1[(pl * NPT + n) * LDF + h]);
    sP1[pl * LDF + h] = mx;
  }
  if (tid < PG) {
    float v = 0.f;
    for (int n = 0; n < NPT; ++n) v = fmaxf(v, sMask[tid * NPT + n]);
    sValid[tid] = v;
  }
  __syncthreads();

  // ---------- Phase D: layer1, K=128 ([feat | pooled] virtual concat), WMMA ----------
  for (int t = wave; t < 40; t += 8) {            // 10 M-tiles x 4 N-tiles
    int mB = (t >> 2) * 16, nB = (t & 3) * 16;
    int r  = mB + lq;
    int pl = r / NPT;
    const float* aF = sF1 + r * LDF + 2 * half;
    const float* aP = sP1 + pl * LDF + 2 * half;
    const float* bW = sW + (nB + lq) * LDW1 + 2 * half;
    v8f acc = {0.f, 0.f, 0.f, 0.f, 0.f, 0.f, 0.f, 0.f};
#pragma unroll
    for (int k0 = 0; k0 < 2 * HH; k0 += 4) {
      v2f a = (k0 < HH) ? *(const v2f*)(aF + k0) : *(const v2f*)(aP + (k0 - HH));
      v2f b = *(const v2f*)(bW + k0);
      acc = wmma_f32(a, b, acc);
    }
#pragma unroll
    for (int i = 0; i < 8; ++i) {                 // D layout: VGPR i -> M = i + 8*half
      int row = mB + i + 8 * half, n = nB + lq;
      sF2[row * LDF + n] = fmaxf(acc[i] * sSc1[n] + sSh1[n], 0.f);
    }
  }
  __syncthreads();

  // ---------- Phase E: stage W2 (64x64) ----------
  for (int e = tid; e < HH * HH; e += 256) sW[(e >> 6) * LDF + (e & 63)] = W2[e];
  __syncthreads();

  // ---------- Phase F: layer2, K=64, WMMA; output masked ----------
  for (int t = wave; t < 40; t += 8) {
    int mB = (t >> 2) * 16, nB = (t & 3) * 16;
    int r  = mB + lq;
    const float* aF = sF2 + r * LDF + 2 * half;
    const float* bW = sW + (nB + lq) * LDF + 2 * half;
    v8f acc = {0.f, 0.f, 0.f, 0.f, 0.f, 0.f, 0.f, 0.f};
#pragma unroll
    for (int k0 = 0; k0 < HH; k0 += 4) {
      v2f a = *(const v2f*)(aF + k0);
      v2f b = *(const v2f*)(bW + k0);
      acc = wmma_f32(a, b, acc);
    }
#pragma unroll
    for (int i = 0; i < 8; ++i) {
      int row = mB + i + 8 * half, n = nB + lq;
      sF1[row * LDF + n] =
          fmaxf(acc[i] * sSc2[n] + sSh2[n], 0.f) * sMask[row];
    }
  }
  __syncthreads();

  // ---------- Phase G: second max-pool (pad to 16 rows) + stage Wo1 ----------
  for (int o = tid; o < 16 * HH; o += 256) {
    int pl = o >> 6, h = o & 63;
    float mx = 0.f;
    if (pl < PG)
      for (int n = 0; n < NPT; ++n) mx = fmaxf(mx, sF1[(pl * NPT + n) * LDF + h]);
    sP2[pl * LDF + h] = mx;
  }
  for (int e = tid; e < HH * HH; e += 256) sW[(e >> 6) * LDF + (e & 63)] = Wo1[e];
  __syncthreads();

  // ---------- Phase H: head hidden = relu(pooled2 @ Wo1.T + bo1), 1x4 tiles ----------
  if (wave < 4) {                                 // wave-uniform: EXEC stays all-ones
    int nB = wave * 16;
    const float* aF = sP2 + lq * LDF + 2 * half;
    const float* bW = sW + (nB + lq) * LDF + 2 * half;
    v8f acc = {0.f, 0.f, 0.f, 0.f, 0.f, 0.f, 0.f, 0.f};
#pragma unroll
    for (int k0 = 0; k0 < HH; k0 += 4) {
      v2f a = *(const v2f*)(aF + k0);
      v2f b = *(const v2f*)(bW + k0);
      acc = wmma_f32(a, b, acc);
    }
#pragma unroll
    for (int i = 0; i < 8; ++i) {
      int row = i + 8 * half, n = nB + lq;
      sHid[row * LDF + n] = fmaxf(acc[i] + sBo1[n], 0.f);
    }
  }
  __syncthreads();

  // ---------- Phase I: stage Wo2 (128x64) ----------
  for (int e = tid; e < OO * HH; e += 256) sW[(e >> 6) * LDF + (e & 63)] = Wo2[e];
  __syncthreads();

  // ---------- Phase J: out = hidden @ Wo2.T + bo2, valid-masked, 1x8 tiles ----------
  {
    int nB = wave * 16;                           // 8 waves cover O=128
    const float* aF = sHid + lq * LDF + 2 * half;
    const float* bW = sW + (nB + lq) * LDF + 2 * half;
    v8f acc = {0.f, 0.f, 0.f, 0.f, 0.f, 0.f, 0.f, 0.f};
#pragma unroll
    for (int k0 = 0; k0 < HH; k0 += 4) {
      v2f a = *(const v2f*)(aF + k0);
      v2f b = *(const v2f*)(bW + k0);
      acc = wmma_f32(a, b, acc);
    }
    if (half == 0) {                              // rows 0..7 are the real polylines
      int n = nB + lq;
#pragma unroll
      for (int i = 0; i < PG; ++i)
        out[(size_t)(q0 + i) * OO + n] = (acc[i] + sBo2[n]) * sValid[i];
    }
  }
}

extern "C" void kernel_launch(void* const* d_in, const int* in_sizes, int n_in,
                              void* d_out, int out_size, void* d_ws, size_t ws_size,
                              hipStream_t stream) {
  (void)in_sizes; (void)n_in; (void)out_size; (void)d_ws; (void)ws_size;
  const float* poly = (const float*)d_in[0];
  const unsigned char* maskb = (const unsigned char*)d_in[1];
  const float* W0 = (const float*)d_in[2];
  const float* g0 = (const float*)d_in[3];
  const float* b0 = (const float*)d_in[4];
  const float* m0 = (const float*)d_in[5];
  const float* v0 = (const float*)d_in[6];
  const float* W1 = (const float*)d_in[7];
  const float* g1 = (const float*)d_in[8];
  const float* b1 = (const float*)d_in[9];
  const float* m1 = (const float*)d_in[10];
  const float* v1 = (const float*)d_in[11];
  const float* W2 = (const float*)d_in[12];
  const float* g2 = (const float*)d_in[13];
  const float* b2 = (const float*)d_in[14];
  const float* m2 = (const float*)d_in[15];
  const float* v2 = (const float*)d_in[16];
  const float* Wo1 = (const float*)d_in[17];
  const float* bo1 = (const float*)d_in[18];
  const float* Wo2 = (const float*)d_in[19];
  const float* bo2 = (const float*)d_in[20];
  float* out = (float*)d_out;

  dim3 grid((BB * PP) / PG);   // 3072 workgroups
  dim3 block(256);             // 8 waves (wave32)
  size_t shmem = (size_t)SMEM_FLOATS * sizeof(float);   // ~138 KB of the 320 KB WGP LDS
  pointnet_poly_kernel<<<grid, block, shmem, stream>>>(
      poly, maskb, W0, g0, b0, m0, v0, W1, g1, b1, m1, v1,
      W2, g2, b2, m2, v2, Wo1, bo1, Wo2, bo2, out);
}